// Gconv_last_55413668053695
// MI455X (gfx1250) — compile-verified
//
#include <hip/hip_runtime.h>

// CDNA5 / gfx1250. wave32. Matrix op: V_WMMA_F32_16X16X4_F32 (exact f32 path;
// x_new is a directly-validated f32 output over K=1024, so low-precision WMMA
// is too risky). See analysis: fused P-trick turns the whole gconv into one
// streaming pass over W and y.

typedef __attribute__((ext_vector_type(2))) float v2f;
typedef __attribute__((ext_vector_type(8))) float v8f;

#define NDIM 1024
#define BS   16
#define NB   256          // NDIM / 4 K-steps
#define PFRAG_FLOATS (BS * NB * 3 * 4 * 64)   // 3,145,728 floats = 12.58 MB

// ---------------------------------------------------------------------------
// Kernel A: P_src[b,n,f] = sum_c x[b,n,c] * fc_w[f, src*64 + c],
// written directly in WMMA B-fragment order:
//   pfrag[(((b*NB+nb)*3+src)*4+fc)*64 + lane*2 + vg]
//     = P_src[b, nb*4 + 2*(lane>=16) + vg, fc*16 + (lane&15)]
// ---------------------------------------------------------------------------
__global__ void __launch_bounds__(256) pfrag_kernel(
    const float* __restrict__ x, const float* __restrict__ fc_w,
    float* __restrict__ pfrag) {
  unsigned tid    = blockIdx.x * 256u + threadIdx.x;   // [0, PFRAG_FLOATS)
  unsigned within = tid & 63u;
  unsigned lane   = within >> 1;
  unsigned vg     = within & 1u;
  unsigned g      = tid >> 6;
  unsigned fc     = g & 3u;
  unsigned g2     = g >> 2;
  unsigned src    = g2 % 3u;
  unsigned g3     = g2 / 3u;
  unsigned nb     = g3 & 255u;
  unsigned b      = g3 >> 8;

  unsigned n = nb * 4u + ((lane >= 16u) ? 2u : 0u) + vg;
  unsigned f = fc * 16u + (lane & 15u);

  const float* xr = x + ((size_t)b * NDIM + n) * 64;
  const float* wr = fc_w + (size_t)f * 192 + src * 64;
  float s = 0.0f;
#pragma unroll
  for (int c = 0; c < 64; ++c) s = fmaf(xr[c], wr[c], s);
  pfrag[tid] = s;
}

// ---------------------------------------------------------------------------
// Kernel B: main fused matmul.  One wave = one [16 x 64] x_new tile.
//   x_new[b,m] = sum_n W0[m,n]*P0[n] + W1[m,n]*P1[n] + y[m,n]*P2[n] + fc_b
// 768 WMMA K-steps of 4, 12 v_wmma_f32_16x16x4_f32 per nb iteration.
// Epilogue: bx[b,m] = x_new[b,m].beta_w + beta_b via half-wave shfl reduce.
// ---------------------------------------------------------------------------
__global__ void __launch_bounds__(128) gconv_main(
    const float* __restrict__ W, const float* __restrict__ y,
    const float* __restrict__ pfrag, const float* __restrict__ fc_b,
    const float* __restrict__ beta_w, const float* __restrict__ beta_b,
    float* __restrict__ xnew, float* __restrict__ bx_out) {
  const int lane = threadIdx.x & 31;
  const int wave = threadIdx.x >> 5;
  const int tile = blockIdx.x * 4 + wave;      // 1024 tiles total
  const int b    = tile >> 6;
  const int m0   = (tile & 63) << 4;
  const int m    = lane & 15;
  const int hi   = lane >> 4;                  // half-wave select
  const int klo  = hi << 1;                    // k offset 0 or 2

  // A-side pointers: W row gives {W0[n],W1[n],W0[n+1],W1[n+1]} as one b128.
  const float4* wrow = (const float4*)(W + (((size_t)(b * NDIM + m0 + m)) * NDIM + klo) * 2);
  const float2* yrow = (const float2*)(y + ((size_t)(b * NDIM + m0 + m)) * NDIM + klo);
  // B-side fragments (pre-swizzled by pfrag_kernel): one float2 per lane.
  const float2* pbase = (const float2*)pfrag + (size_t)b * (NB * 3 * 4 * 32) + lane;

  v8f acc[4] = {};   // 4 column tiles of 16x16 f32

  for (int nb = 0; nb < NB; ++nb) {
    float4 wv = wrow[(size_t)nb * 2];
    float2 yv = yrow[(size_t)nb * 2];
    v2f A[3];
    A[0] = v2f{wv.x, wv.z};      // W0: k={klo, klo+1}
    A[1] = v2f{wv.y, wv.w};      // W1
    A[2] = v2f{yv.x, yv.y};      // y
    const float2* p = pbase + (size_t)nb * 384;   // 3*4*32 float2 per nb
#pragma unroll
    for (int src = 0; src < 3; ++src) {
#pragma unroll
      for (int fc = 0; fc < 4; ++fc) {
        float2 bv = p[(src * 4 + fc) * 32];
        v2f Bf = v2f{bv.x, bv.y};
        acc[fc] = __builtin_amdgcn_wmma_f32_16x16x4_f32(
            false, A[src], false, Bf, (short)0, acc[fc], false, false);
      }
    }
  }

  // ---- epilogue ----
  const int nn = lane & 15;      // column within 16-wide tile
  // add fc bias (column-indexed => same value for all 8 row-VGPRs)
#pragma unroll
  for (int fc = 0; fc < 4; ++fc) {
    float fb = fc_b[fc * 16 + nn];
#pragma unroll
    for (int r = 0; r < 8; ++r) acc[fc][r] += fb;
  }

  // store x_new  (C layout: VGPR r, halfwave hi -> row r+8*hi, col = lane%16)
  float* xo = xnew + ((size_t)(b * NDIM + m0)) * 64;
#pragma unroll
  for (int fc = 0; fc < 4; ++fc) {
#pragma unroll
    for (int r = 0; r < 8; ++r) {
      int mm = r + 8 * hi;
      xo[(size_t)mm * 64 + fc * 16 + nn] = acc[fc][r];
    }
  }

  // bx[m] = sum_f x_new[m,f] * beta_w[f] + beta_b ; reduce 16 lanes per half.
  float pr[8];
#pragma unroll
  for (int r = 0; r < 8; ++r) {
    float s = 0.0f;
#pragma unroll
    for (int fc = 0; fc < 4; ++fc) s = fmaf(acc[fc][r], beta_w[fc * 16 + nn], s);
    s += __shfl_xor(s, 1, 32);
    s += __shfl_xor(s, 2, 32);
    s += __shfl_xor(s, 4, 32);
    s += __shfl_xor(s, 8, 32);   // masks < 16: stays within each half-wave
    pr[r] = s;
  }
  if (nn == 0) {
    float bb = beta_b[0];
#pragma unroll
    for (int r = 0; r < 8; ++r)
      bx_out[b * NDIM + m0 + r + 8 * hi] = pr[r] + bb;
  }
}

// ---------------------------------------------------------------------------
// Kernel C: y_new[b,m,n] = sigmoid(bx[b,m]+bx[b,n]+sigma*y) * (m!=n)
// ---------------------------------------------------------------------------
__global__ void __launch_bounds__(256) gate_kernel(
    const float* __restrict__ y, const float* __restrict__ bx,
    const float* __restrict__ sigma, float* __restrict__ ynew) {
  unsigned idx = blockIdx.x * 256u + threadIdx.x;   // [0, 16*1024*1024)
  unsigned n = idx & 1023u;
  unsigned m = (idx >> 10) & 1023u;
  unsigned b = idx >> 20;
  float z = bx[(b << 10) + m] + bx[(b << 10) + n] + sigma[0] * y[idx];
  float g = 1.0f / (1.0f + __expf(-z));
  ynew[idx] = (m == n) ? 0.0f : g;
}

// ---------------------------------------------------------------------------
extern "C" void kernel_launch(void* const* d_in, const int* in_sizes, int n_in,
                              void* d_out, int out_size, void* d_ws, size_t ws_size,
                              hipStream_t stream) {
  const float* W      = (const float*)d_in[0];   // [16,1024,1024,2]
  const float* x      = (const float*)d_in[1];   // [16,1024,64]
  const float* y      = (const float*)d_in[2];   // [16,1024,1024]
  const float* fc_w   = (const float*)d_in[3];   // [64,192]
  const float* fc_b   = (const float*)d_in[4];   // [64]
  const float* beta_w = (const float*)d_in[5];   // [1,64]
  const float* beta_b = (const float*)d_in[6];   // [1]
  const float* sigma  = (const float*)d_in[7];   // [1]

  float* out   = (float*)d_out;
  float* W_out = out;                                   // 33,554,432 floats
  float* xnew  = out + 33554432;                        //  1,048,576 floats
  float* ynew  = out + 33554432 + 1048576;              // 16,777,216 floats

  float* pfrag = (float*)d_ws;                          // 12.58 MB
  float* bx    = pfrag + PFRAG_FLOATS;                  // 64 KB

  // W passes through unchanged: pure d2d stream copy.
  hipMemcpyAsync(W_out, W, (size_t)33554432 * sizeof(float),
                 hipMemcpyDeviceToDevice, stream);

  pfrag_kernel<<<PFRAG_FLOATS / 256, 256, 0, stream>>>(x, fc_w, pfrag);
  gconv_main<<<256, 128, 0, stream>>>(W, y, pfrag, fc_b, beta_w, beta_b, xnew, bx);
  gate_kernel<<<(16u * 1024u * 1024u) / 256u, 256, 0, stream>>>(y, bx, sigma, ynew);
}